// CrossAttention_Sp_49014166782306
// MI455X (gfx1250) — compile-verified
//
#include <hip/hip_runtime.h>

#define DIMD 256
#define NTOK 4096
#define NWAVES 8

typedef __attribute__((ext_vector_type(2))) float v2f;
typedef __attribute__((ext_vector_type(8))) float v8f;

__device__ inline v8f wmma4(v2f a, v2f b, v8f c) {
  // D = A(16x4) * B(4x16) + C, full f32 (CDNA5 V_WMMA_F32_16X16X4_F32)
  return __builtin_amdgcn_wmma_f32_16x16x4_f32(false, a, false, b, (short)0, c,
                                               false, false);
}

__device__ inline void om_merge(float& m, float& s, float m2, float s2) {
  float nm = fmaxf(m, m2);
  s = s * __expf(m - nm) + s2 * __expf(m2 - nm);
  m = nm;
}

// ---------------- Kernel 1: W_c = U^T diag(S_c^2) U  (symmetric, 2x256x256)
__global__ void k_wc(const float* __restrict__ U, const float* __restrict__ S1,
                     const float* __restrict__ S2, float* __restrict__ wc) {
  int idx = blockIdx.x * blockDim.x + threadIdx.x;  // 131072 threads
  int c = idx >> 16;
  int ij = idx & 65535;
  int i = ij >> 8;
  int j = ij & 255;
  const float* S = c ? S2 : S1;
  float acc = 0.f;
  for (int k = 0; k < DIMD; ++k) {
    float s = S[k];
    acc += U[k * DIMD + i] * (s * s) * U[k * DIMD + j];
  }
  wc[idx] = acc;
}

// ---------------- Kernel 2: xs_c = x @ W_c  (W_c symmetric -> use B^T pattern)
__global__ void __launch_bounds__(256) k_xs(const float* __restrict__ x,
                                            const float* __restrict__ wc,
                                            float* __restrict__ xs) {
  __shared__ __attribute__((aligned(16))) float ldsA[16 * DIMD];
  int tile = blockIdx.x;  // 0..511 (16 rows of B*N=8192 each)
  int c = blockIdx.y;     // branch
  int tid = threadIdx.x;
  int lane = tid & 31, wave = tid >> 5;
  int ln = lane & 15, hh = lane >> 4;
  int g0 = tile * 16;
  int b = g0 >> 12;       // /4096
  int n0 = g0 & 4095;

  for (int i = tid; i < 16 * DIMD; i += 256) ldsA[i] = x[(size_t)g0 * DIMD + i];
  __syncthreads();

  const float* wcc = wc + (size_t)c * DIMD * DIMD;
  size_t obase = ((size_t)(b * 2 + c) * NTOK + n0) * DIMD;
  for (int jj = 0; jj < 2; ++jj) {
    int j0 = (wave * 2 + jj) * 16;
    v8f acc = {};
#pragma unroll
    for (int kk = 0; kk < 64; ++kk) {
      v2f a = *(const v2f*)&ldsA[ln * DIMD + kk * 4 + 2 * hh];
      v2f bb = *(const v2f*)&wcc[(size_t)(j0 + ln) * DIMD + kk * 4 + 2 * hh];
      acc = wmma4(a, bb, acc);
    }
#pragma unroll
    for (int v = 0; v < 8; ++v)
      xs[obase + (size_t)(v + 8 * hh) * DIMD + j0 + ln] = acc[v];
  }
}

// ---------------- Kernel 3: fused attention (flash-style, 2 passes)
// One block per (b, c, 16-row tile). 8 waves; each wave owns 32 of the 256
// m-tiles. Pass 1: WMMA logits + online softmax stats. Pass 2: recompute
// logits, a = (1-g)p + g q, entropy, and attn@y via a second WMMA GEMM
// (a-tile transposed through LDS into A-layout).
__global__ void __launch_bounds__(256) k_attn(const float* __restrict__ xs,
                                              const float* __restrict__ y,
                                              const float* __restrict__ focusp,
                                              const float* __restrict__ gatingp,
                                              float* __restrict__ xo,
                                              float* __restrict__ entw) {
  __shared__ __attribute__((aligned(16))) float ldsA[16 * DIMD];   // xs tile
  __shared__ __attribute__((aligned(16))) float ldsOut[16 * DIMD]; // out accum
  __shared__ __attribute__((aligned(16))) float ldsT[NWAVES * 256]; // transpose
  __shared__ float ldsQ[64];           // positional score per ay_m
  __shared__ float ldsStats[16 * 2];   // (rowmax, 1/rowsum)
  __shared__ float ldsRed[NWAVES * 16 * 2];
  __shared__ float ldsEnt[16];

  int tile = blockIdx.x;  // 0..255
  int c = blockIdx.y;
  int b = blockIdx.z;
  int tid = threadIdx.x;
  int lane = tid & 31, wave = tid >> 5;
  int ln = lane & 15, hh = lane >> 4;
  int n0 = tile * 16;

  float f = fabsf(focusp[0]);
  float g = 1.f / (1.f + __expf(-gatingp[0]));
  float omg = 1.f - g;

  size_t xsbase = ((size_t)(b * 2 + c) * NTOK + n0) * DIMD;
  const float* yb = y + (size_t)b * NTOK * DIMD;

  for (int i = tid; i < 16 * DIMD; i += 256) {
    ldsA[i] = xs[xsbase + i];
    ldsOut[i] = 0.f;
  }
  if (tid < 16) ldsEnt[tid] = 0.f;
  if (tid < 64) {
    // rel[n,m] = ay_m - ay_n ; ay_n constant over the 16-row tile.
    int ayn = n0 >> 6;
    float den = 0.f;
    for (int rr = 0; rr < 64; ++rr) {
      float d = (float)(rr - ayn);
      den += __expf(-f * d * d);
    }
    den *= 64.f;  // each ay value occurs W=64 times
    float dm = (float)(tid - ayn);
    ldsQ[tid] = __expf(-f * dm * dm) / den;
  }
  __syncthreads();

  // ---- pass 1: online softmax stats over content logits
  float rmax[8], rsum[8];
#pragma unroll
  for (int v = 0; v < 8; ++v) { rmax[v] = -3.4e38f; rsum[v] = 0.f; }
  for (int t = 0; t < 32; ++t) {
    int m0 = (wave + NWAVES * t) * 16;
    v8f acc = {};
#pragma unroll
    for (int kk = 0; kk < 64; ++kk) {
      v2f a = *(const v2f*)&ldsA[ln * DIMD + kk * 4 + 2 * hh];
      v2f bb = *(const v2f*)&yb[(size_t)(m0 + ln) * DIMD + kk * 4 + 2 * hh];
      acc = wmma4(a, bb, acc);
    }
#pragma unroll
    for (int v = 0; v < 8; ++v) {
      float s = acc[v] * 0.0625f;  // d^-0.5 = 1/16
      float nm = fmaxf(rmax[v], s);
      rsum[v] = rsum[v] * __expf(rmax[v] - nm) + __expf(s - nm);
      rmax[v] = nm;
    }
  }
#pragma unroll
  for (int v = 0; v < 8; ++v) {
#pragma unroll
    for (int off = 1; off < 16; off <<= 1) {
      float m2 = __shfl_xor(rmax[v], off, 32);
      float s2 = __shfl_xor(rsum[v], off, 32);
      om_merge(rmax[v], rsum[v], m2, s2);
    }
  }
  if (ln == 0) {
#pragma unroll
    for (int v = 0; v < 8; ++v) {
      int row = v + 8 * hh;
      ldsRed[(wave * 16 + row) * 2 + 0] = rmax[v];
      ldsRed[(wave * 16 + row) * 2 + 1] = rsum[v];
    }
  }
  __syncthreads();
  if (tid < 16) {
    float M = ldsRed[tid * 2], S = ldsRed[tid * 2 + 1];
    for (int w = 1; w < NWAVES; ++w)
      om_merge(M, S, ldsRed[(w * 16 + tid) * 2], ldsRed[(w * 16 + tid) * 2 + 1]);
    ldsStats[tid * 2] = M;
    ldsStats[tid * 2 + 1] = 1.f / S;
  }
  __syncthreads();

  // ---- pass 2: attn, entropy, and attn @ y
  float Mn[8], Si[8], entp[8];
#pragma unroll
  for (int v = 0; v < 8; ++v) {
    int row = v + 8 * hh;
    Mn[v] = ldsStats[row * 2];
    Si[v] = ldsStats[row * 2 + 1];
    entp[v] = 0.f;
  }
  v8f oacc[16];
  {
    v8f z = {};
#pragma unroll
    for (int dt = 0; dt < 16; ++dt) oacc[dt] = z;
  }
  float* tbuf = &ldsT[wave * 256];
  for (int t = 0; t < 32; ++t) {
    int m0 = (wave + NWAVES * t) * 16;
    float qg = g * ldsQ[m0 >> 6];  // ay_m constant within m-tile
    v8f acc = {};
#pragma unroll
    for (int kk = 0; kk < 64; ++kk) {
      v2f a = *(const v2f*)&ldsA[ln * DIMD + kk * 4 + 2 * hh];
      v2f bb = *(const v2f*)&yb[(size_t)(m0 + ln) * DIMD + kk * 4 + 2 * hh];
      acc = wmma4(a, bb, acc);
    }
#pragma unroll
    for (int v = 0; v < 8; ++v) {
      float s = acc[v] * 0.0625f;
      float p = __expf(s - Mn[v]) * Si[v];
      float a = omg * p + qg;               // sums to 1 over m (exact math)
      entp[v] -= a * __logf(a + 1e-8f);
      tbuf[(v + 8 * hh) * 16 + ln] = a;     // D-layout -> (row, col) in LDS
    }
    asm volatile("s_wait_dscnt 0" ::: "memory");  // same-wave LDS RAW
    v2f a2[4];
#pragma unroll
    for (int kk2 = 0; kk2 < 4; ++kk2)  // re-read in WMMA A-layout
      a2[kk2] = *(const v2f*)&tbuf[ln * 16 + kk2 * 4 + 2 * hh];
#pragma unroll
    for (int dt = 0; dt < 16; ++dt) {
#pragma unroll
      for (int kk2 = 0; kk2 < 4; ++kk2) {
        int krow = m0 + kk2 * 4 + 2 * hh;
        v2f bb;
        bb.x = yb[(size_t)krow * DIMD + dt * 16 + ln];
        bb.y = yb[(size_t)(krow + 1) * DIMD + dt * 16 + ln];
        oacc[dt] = wmma4(a2[kk2], bb, oacc[dt]);
      }
    }
  }

  // entropy: sum across the 16 lanes covering each row, then across waves
#pragma unroll
  for (int v = 0; v < 8; ++v) {
#pragma unroll
    for (int off = 1; off < 16; off <<= 1)
      entp[v] += __shfl_xor(entp[v], off, 32);
  }
  if (ln == 0) {
#pragma unroll
    for (int v = 0; v < 8; ++v) atomicAdd(&ldsEnt[v + 8 * hh], entp[v]);
  }
  // out: reduce 8 waves' partial tiles
#pragma unroll
  for (int dt = 0; dt < 16; ++dt) {
#pragma unroll
    for (int v = 0; v < 8; ++v)
      atomicAdd(&ldsOut[(v + 8 * hh) * DIMD + dt * 16 + ln], oacc[dt][v]);
  }
  __syncthreads();

  size_t obase = ((size_t)(b * 2 + c) * NTOK + n0) * DIMD;
  for (int i = tid; i < 16 * DIMD; i += 256) xo[obase + i] = ldsOut[i];
  if (tid < 16) entw[(size_t)(b * 2 + c) * NTOK + n0 + tid] = ldsEnt[tid];
}

// ---------------- Kernel 4: routing / selection
__global__ void k_route(const float* __restrict__ xo,
                        const float* __restrict__ entw,
                        const float* __restrict__ tempp,
                        float* __restrict__ out, float* __restrict__ heat) {
  size_t idx = (size_t)blockIdx.x * blockDim.x + threadIdx.x;  // B*N*D
  int k = (int)(idx & 255);
  size_t bn = idx >> 8;  // 0..8191
  int b = (int)(bn >> 12);
  int n = (int)(bn & 4095);
  float T = tempp[0];
  float e0 = entw[(size_t)(b * 2 + 0) * NTOK + n];
  float e1 = entw[(size_t)(b * 2 + 1) * NTOK + n];
  float h0 = 2.f - 2.f / (1.f + __expf(-T * e0));
  float h1 = 2.f - 2.f / (1.f + __expf(-T * e1));
  int c = (h0 >= h1) ? 0 : 1;
  out[idx] = xo[((size_t)(b * 2 + c) * NTOK + n) * DIMD + k];
  if (k == 0) heat[bn] = (h0 >= h1) ? h0 : h1;
}

extern "C" void kernel_launch(void* const* d_in, const int* in_sizes, int n_in,
                              void* d_out, int out_size, void* d_ws,
                              size_t ws_size, hipStream_t stream) {
  const float* x = (const float*)d_in[0];
  const float* y = (const float*)d_in[1];
  const float* U = (const float*)d_in[2];
  const float* S1 = (const float*)d_in[3];
  const float* S2 = (const float*)d_in[4];
  const float* focus = (const float*)d_in[5];
  const float* gating = (const float*)d_in[6];
  const float* temp = (const float*)d_in[7];
  // d_in[8] (rel_coords_y, 64 MB) is recomputed analytically: never loaded.

  float* ws = (float*)d_ws;
  float* wc = ws;                    // 2*256*256        = 131072
  float* xs = wc + 131072;           // B*2*N*D          = 4194304
  float* xo = xs + 4194304;          // B*2*N*D          = 4194304
  float* ent = xo + 4194304;         // B*2*N            = 16384
  float* out = (float*)d_out;        // B*N*D
  float* heat = out + (size_t)2 * NTOK * DIMD;  // B*N

  k_wc<<<512, 256, 0, stream>>>(U, S1, S2, wc);
  dim3 gxs(512, 2);
  k_xs<<<gxs, 256, 0, stream>>>(x, wc, xs);
  dim3 gat(256, 2, 2);
  k_attn<<<gat, 256, 0, stream>>>(xs, y, focus, gating, xo, ent);
  k_route<<<8192, 256, 0, stream>>>(xo, ent, temp, out, heat);
}